// DyIGCN_89575837926027
// MI455X (gfx1250) — compile-verified
//
#include <hip/hip_runtime.h>
#include <math.h>

// ---------------------------------------------------------------------------
// DyIGCN forward on MI455X (gfx1250, wave32).
//   out_final [N x 2]  = relu(zbar @ Wfc1 + b1) @ Wfc2 + b2
//   zbar      [N x 16] = GRU(relu(GCNConv(x)), Zt)       (first==0 path)
// d_out = [out_final | zbar] concatenated (fp32).
//
// Roofline: edge phase dominates (~600MB HBM). Edge-index streams are marked
// non-temporal so the 128MB xs/acc random-access working set stays resident
// in the 192MB L2 during the 256M scatter-atomics. Dense per-node math uses
// V_WMMA_F32_16X16X4_F32 on 16-node tiles (full fp32 precision).
// ---------------------------------------------------------------------------

typedef float v2f __attribute__((ext_vector_type(2)));
typedef float v8f __attribute__((ext_vector_type(8)));

#define FEAT 16

__device__ __forceinline__ v8f wmma_f32_k4(v2f a, v2f b, v8f c) {
    // D = A(16x4) x B(4x16) + C(16x16), fp32
    return __builtin_amdgcn_wmma_f32_16x16x4_f32(false, a, false, b,
                                                 (short)0, c, false, false);
}

// Full K=16 matmul: C += A(16x16) @ W(16x16), A pre-loaded in A-layout slices.
__device__ __forceinline__ v8f mm16(const v2f a[4], const float* __restrict__ W,
                                    int h, int col, v8f c) {
#pragma unroll
    for (int s = 0; s < 4; ++s) {
        int r0 = 4 * s + 2 * h;
        v2f b;
        b[0] = W[r0 * FEAT + col];
        b[1] = W[(r0 + 1) * FEAT + col];
        c = wmma_f32_k4(a[s], b, c);
    }
    return c;
}

__device__ __forceinline__ float sigm(float x) { return 1.0f / (1.0f + __expf(-x)); }

// --------------------------- kernels ---------------------------------------

// deg[i] = 1 (self loop)
__global__ void k_init_deg(float* __restrict__ deg, int n) {
    int i = blockIdx.x * blockDim.x + threadIdx.x;
    if (i < n) deg[i] = 1.0f;
}

// deg[col[e]] += 1   (edge list is a one-shot stream -> NT load)
__global__ void k_degree(const long long* __restrict__ col, float* __restrict__ deg,
                         long long E) {
    long long e = (long long)blockIdx.x * blockDim.x + threadIdx.x;
    if (e < E) {
        long long c = __builtin_nontemporal_load(col + e);
        atomicAdd(&deg[c], 1.0f);
    }
}

// in-place: dis[i] = rsqrt(deg[i])   (deg >= 1 always, no zero guard needed)
__global__ void k_rsqrt(float* __restrict__ deg, int n) {
    int i = blockIdx.x * blockDim.x + threadIdx.x;
    if (i < n) deg[i] = rsqrtf(deg[i]);
}

// Per 16-node tile (one wave): xw = x @ W_conv via WMMA; xs = dis * xw;
// acc (zbar region of d_out) initialized to xs (self-loop term).
__global__ void k_xw_wmma(const float* __restrict__ x, const float* __restrict__ Wc,
                          const float* __restrict__ dis, float* __restrict__ xs,
                          float* __restrict__ acc, int n) {
    int wave = (int)((blockIdx.x * blockDim.x + threadIdx.x) >> 5);
    int lane = threadIdx.x & 31;
    int base = wave * 16;
    if (base >= n) return;                  // wave-uniform: EXEC stays all-1s
    int h = lane >> 4;
    int m = lane & 15;                      // A-layout row / B,C-layout column

    int nodeA = base + m; if (nodeA >= n) nodeA = n - 1;
    v2f a[4];
#pragma unroll
    for (int s = 0; s < 4; ++s)             // x is read exactly once -> NT
        a[s] = __builtin_nontemporal_load(
            reinterpret_cast<const v2f*>(x + (size_t)nodeA * FEAT + 4 * s + 2 * h));

    v8f c = {};
    c = mm16(a, Wc, h, m, c);

    if (base + 16 <= n) {                   // full tile: unguarded clause stores
#pragma unroll
        for (int vr = 0; vr < 8; ++vr) {
            int node = base + vr + 8 * h;
            float v = c[vr] * dis[node];
            xs[(size_t)node * FEAT + m]  = v;
            acc[(size_t)node * FEAT + m] = v;
        }
    } else {                                // last partial tile only
#pragma unroll
        for (int vr = 0; vr < 8; ++vr) {
            int node = base + vr + 8 * h;
            if (node < n) {
                float v = c[vr] * dis[node];
                xs[(size_t)node * FEAT + m]  = v;
                acc[(size_t)node * FEAT + m] = v;
            }
        }
    }
}

// One thread per (edge, feature): acc[col*16+k] += xs[row*16+k]
// lanes 0..15 of a 16-group share the edge -> index loads merge into one line,
// atomics of the group hit one cacheline. Edge list NT; xs stays L2-hot.
__global__ void k_edge_scatter(const long long* __restrict__ row,
                               const long long* __restrict__ col,
                               const float* __restrict__ xs,
                               float* __restrict__ acc, long long E) {
    long long t = (long long)blockIdx.x * blockDim.x + threadIdx.x;
    long long e = t >> 4;
    int k = (int)(t & 15);
    if (e < E) {
        long long r  = __builtin_nontemporal_load(row + e);
        long long cI = __builtin_nontemporal_load(col + e);
        atomicAdd(acc + cI * FEAT + k, xs[r * FEAT + k]);
    }
}

// Fused: x_emb = relu(dis*acc + b_conv); zbar = GRU(x_emb, Zt) (or x_emb if first).
// Reads acc tile / writes zbar tile in place (per-wave exclusive 16-node tile).
__global__ void k_gru_wmma(float* __restrict__ zb,          // acc in, zbar out
                           const float* __restrict__ dis,
                           const float* __restrict__ Zt,
                           const int* __restrict__ firstp,
                           const float* __restrict__ bc,
                           const float* __restrict__ Wir, const float* __restrict__ bir,
                           const float* __restrict__ Whr, const float* __restrict__ bhr,
                           const float* __restrict__ Wiz, const float* __restrict__ biz,
                           const float* __restrict__ Whz, const float* __restrict__ bhz,
                           const float* __restrict__ Win, const float* __restrict__ bin_,
                           const float* __restrict__ Whn, const float* __restrict__ bhn,
                           int n) {
    int wave = (int)((blockIdx.x * blockDim.x + threadIdx.x) >> 5);
    int lane = threadIdx.x & 31;
    int base = wave * 16;
    if (base >= n) return;                  // wave-uniform
    int h = lane >> 4;
    int m = lane & 15;
    bool first = (firstp[0] != 0);
    bool full  = (base + 16 <= n);

    if (first) {
        // zbar = relu(dis*acc + b_conv), C-layout in-place
#pragma unroll
        for (int vr = 0; vr < 8; ++vr) {
            int node = base + vr + 8 * h;
            if (full || node < n) {
                size_t idx = (size_t)node * FEAT + m;
                zb[idx] = fmaxf(zb[idx] * dis[node] + bc[m], 0.0f);
            }
        }
        return;
    }

    // A-layout load of x_emb (activation applied element-wise) and Zt.
    int nodeA = base + m; if (nodeA >= n) nodeA = n - 1;
    float dA = dis[nodeA];
    v2f Ae[4], Az[4];
#pragma unroll
    for (int s = 0; s < 4; ++s) {
        int c0 = 4 * s + 2 * h;
        v2f raw = *reinterpret_cast<const v2f*>(zb + (size_t)nodeA * FEAT + c0);
        v2f e;
        e[0] = fmaxf(raw[0] * dA + bc[c0], 0.0f);
        e[1] = fmaxf(raw[1] * dA + bc[c0 + 1], 0.0f);
        Ae[s] = e;
        Az[s] = *reinterpret_cast<const v2f*>(Zt + (size_t)nodeA * FEAT + c0);
    }

    v8f zero = {};
    v8f rpre = mm16(Ae, Wir, h, m, zero);
    rpre     = mm16(Az, Whr, h, m, rpre);
    v8f zpre = mm16(Ae, Wiz, h, m, zero);
    zpre     = mm16(Az, Whz, h, m, zpre);
    v8f ipre = mm16(Ae, Win, h, m, zero);
    v8f hpre = mm16(Az, Whn, h, m, zero);

    float br = bir[m] + bhr[m];
    float bz = biz[m] + bhz[m];
    float bi = bin_[m];
    float bh = bhn[m];

    v8f r, z, nn;
#pragma unroll
    for (int i = 0; i < 8; ++i) {
        r[i] = sigm(rpre[i] + br);
        z[i] = sigm(zpre[i] + bz);
        nn[i] = tanhf(ipre[i] + bi + r[i] * (hpre[i] + bh));
    }

    // blend with Zt (C-layout) and store zbar over acc
    if (full) {
#pragma unroll
        for (int vr = 0; vr < 8; ++vr) {
            size_t idx = (size_t)(base + vr + 8 * h) * FEAT + m;
            float B = Zt[idx];
            zb[idx] = (1.0f - z[vr]) * nn[vr] + z[vr] * B;
        }
    } else {
#pragma unroll
        for (int vr = 0; vr < 8; ++vr) {
            int node = base + vr + 8 * h;
            if (node < n) {
                size_t idx = (size_t)node * FEAT + m;
                float B = Zt[idx];
                zb[idx] = (1.0f - z[vr]) * nn[vr] + z[vr] * B;
            }
        }
    }
}

// FC head: out = relu(zbar @ W1 + b1) @ W2 + b2   (16 -> 8 -> 2, memory bound)
__global__ void k_fc(const float* __restrict__ zbar,
                     const float* __restrict__ W1, const float* __restrict__ b1,
                     const float* __restrict__ W2, const float* __restrict__ b2,
                     float* __restrict__ out, int n) {
    int i = blockIdx.x * blockDim.x + threadIdx.x;
    if (i >= n) return;
    float zv[16];
#pragma unroll
    for (int k = 0; k < 16; ++k) zv[k] = zbar[(size_t)i * FEAT + k];
    float hid[8];
#pragma unroll
    for (int j = 0; j < 8; ++j) {
        float a = b1[j];
#pragma unroll
        for (int k = 0; k < 16; ++k) a = fmaf(zv[k], W1[k * 8 + j], a);
        hid[j] = fmaxf(a, 0.0f);
    }
#pragma unroll
    for (int o = 0; o < 2; ++o) {
        float a = b2[o];
#pragma unroll
        for (int j = 0; j < 8; ++j) a = fmaf(hid[j], W2[j * 2 + o], a);
        out[(size_t)i * 2 + o] = a;
    }
}

// --------------------------- launcher ---------------------------------------

extern "C" void kernel_launch(void* const* d_in, const int* in_sizes, int n_in,
                              void* d_out, int out_size, void* d_ws, size_t ws_size,
                              hipStream_t stream) {
    const float*     x    = (const float*)d_in[0];
    const long long* ei   = (const long long*)d_in[1];   // [2, E] int64
    const float*     Zt   = (const float*)d_in[2];
    const int*       first= (const int*)d_in[3];
    const float*     Wc   = (const float*)d_in[4];
    const float*     bc   = (const float*)d_in[5];
    const float*     Wir  = (const float*)d_in[6];
    const float*     bir  = (const float*)d_in[7];
    const float*     Whr  = (const float*)d_in[8];
    const float*     bhr  = (const float*)d_in[9];
    const float*     Wiz  = (const float*)d_in[10];
    const float*     biz  = (const float*)d_in[11];
    const float*     Whz  = (const float*)d_in[12];
    const float*     bhz  = (const float*)d_in[13];
    const float*     Win  = (const float*)d_in[14];
    const float*     bin_ = (const float*)d_in[15];
    const float*     Whn  = (const float*)d_in[16];
    const float*     bhn  = (const float*)d_in[17];
    const float*     W1   = (const float*)d_in[18];
    const float*     b1   = (const float*)d_in[19];
    const float*     W2   = (const float*)d_in[20];
    const float*     b2   = (const float*)d_in[21];

    const int       N = in_sizes[0] / FEAT;
    const long long E = in_sizes[1] / 2;

    float* out  = (float*)d_out;
    float* zbar = out + (size_t)N * 2;          // zbar output doubles as acc
    float* deg  = (float*)d_ws;                 // N floats; becomes dis in place
    float* xs   = deg + N;                      // N*16 floats (dis-scaled xw)

    const long long* rowIdx = ei;               // edge_index[0] = sources
    const long long* colIdx = ei + E;           // edge_index[1] = targets

    const int TPB = 256;
    dim3 gN((unsigned)((N + TPB - 1) / TPB));
    dim3 gE((unsigned)((E + TPB - 1) / TPB));
    long long totEF = E * FEAT;
    dim3 gEF((unsigned)((totEF + TPB - 1) / TPB));
    int tiles = (N + 15) / 16;                  // 16-node WMMA tiles, 1 wave each
    dim3 gT((unsigned)((tiles + 7) / 8));       // 8 waves per 256-thread block

    k_init_deg    <<<gN,  TPB, 0, stream>>>(deg, N);
    k_degree      <<<gE,  TPB, 0, stream>>>(colIdx, deg, E);
    k_rsqrt       <<<gN,  TPB, 0, stream>>>(deg, N);
    k_xw_wmma     <<<gT,  TPB, 0, stream>>>(x, Wc, deg, xs, zbar, N);
    k_edge_scatter<<<gEF, TPB, 0, stream>>>(rowIdx, colIdx, xs, zbar, E);
    k_gru_wmma    <<<gT,  TPB, 0, stream>>>(zbar, deg, Zt, first, bc,
                                            Wir, bir, Whr, bhr,
                                            Wiz, biz, Whz, bhz,
                                            Win, bin_, Whn, bhn, N);
    k_fc          <<<gN,  TPB, 0, stream>>>(zbar, W1, b1, W2, b2, out, N);
}